// CICDM_Net_42314017800275
// MI455X (gfx1250) — compile-verified
//
#include <hip/hip_runtime.h>
#include <hip/hip_bf16.h>
#include <math.h>

// ---------------------------------------------------------------------------
// CICDM on MI455X (gfx1250, wave32).
// All GEMMs use V_WMMA_F32_16X16X4_F32 (exact f32 matrix path; the workload
// is gather/latency bound, ~1.5 GFLOP total, so low-precision WMMA buys
// nothing). GEMM waves are register-blocked 16x64 (4 accumulators, shared
// A-fragment) -> 1.25 fragment loads per WMMA instead of 2.
// ---------------------------------------------------------------------------

#define E_NUM 4096
#define C_NUM 512
#define P_NUM 256
#define HDIM  128
#define NSTU  128
#define LSEQ  512
#define NEGV  (-1e9f)

typedef __attribute__((ext_vector_type(2))) float v2f;
typedef __attribute__((ext_vector_type(8))) float v8f;

__device__ __forceinline__ v8f wmma4(v2f a, v2f b, v8f c) {
  // D = A(16x4) x B(4x16) + C, f32, wave32
  return __builtin_amdgcn_wmma_f32_16x16x4_f32(false, a, false, b, (short)0, c,
                                               false, false);
}

__device__ __forceinline__ float sigm(float x) {
  return 1.0f / (1.0f + __expf(-x));
}

// -------------------------------------------------------------------- norms
// one wave per row; H=128 -> each lane loads a float4
__global__ void rnorm_kernel(const float* __restrict__ X, float* __restrict__ inv,
                             int rows) {
  int wid  = (blockIdx.x * blockDim.x + threadIdx.x) >> 5;
  int lane = threadIdx.x & 31;
  if (wid >= rows) return;
  const float4* p = (const float4*)(X + (size_t)wid * HDIM);
  float4 v = p[lane];
  float s = v.x * v.x + v.y * v.y + v.z * v.z + v.w * v.w;
  #pragma unroll
  for (int o = 16; o; o >>= 1) s += __shfl_down(s, o, 32);
  if (lane == 0) inv[wid] = rsqrtf(s);
}

// ----------------------------------------------------- cosine sim via WMMA
// S = (X @ Y^T) * invx * invy * 10 ; optional adj mask -> NEG
// one wave per 16x64 output strip (4 accumulators share the A fragment)
__global__ void sim_kernel(const float* __restrict__ X, const float* __restrict__ Y,
                           const float* __restrict__ invx, const float* __restrict__ invy,
                           const float* __restrict__ adj, float* __restrict__ out,
                           int M, int Nc) {
  int wid      = (blockIdx.x * blockDim.x + threadIdx.x) >> 5;
  int tn_count = Nc >> 6;                       // 64-wide column strips
  int tm = wid / tn_count, tn = wid % tn_count;
  if (tm >= (M >> 4)) return;
  int lane = threadIdx.x & 31;
  int half = lane >> 4, r = lane & 15;
  const float* xa = X + (size_t)(tm * 16 + r) * HDIM + 2 * half;
  const float* yb[4];
  #pragma unroll
  for (int j = 0; j < 4; ++j)
    yb[j] = Y + (size_t)(tn * 64 + j * 16 + r) * HDIM + 2 * half;
  v8f acc[4] = {};
  #pragma unroll
  for (int k = 0; k < HDIM; k += 4) {
    v2f a = *(const v2f*)(xa + k);
    #pragma unroll
    for (int j = 0; j < 4; ++j) {
      v2f b = *(const v2f*)(yb[j] + k);
      acc[j] = wmma4(a, b, acc[j]);
    }
  }
  int n = lane & 15, mo = half * 8;
  #pragma unroll
  for (int j = 0; j < 4; ++j) {
    int col = tn * 64 + j * 16 + n;
    float sy = invy[col] * 10.0f;
    #pragma unroll
    for (int v = 0; v < 8; ++v) {
      int row = tm * 16 + mo + v;
      float val = acc[j][v] * invx[row] * sy;
      if (adj && adj[(size_t)row * Nc + col] == 0.0f) val = NEGV;
      out[(size_t)row * Nc + col] = val;
    }
  }
}

// ------------------------------------------------------------- row softmax
__global__ void rowsoftmax_kernel(const float* __restrict__ S, float* __restrict__ out,
                                  int Nc) {
  int row = blockIdx.x;
  const float* p = S + (size_t)row * Nc;
  __shared__ float red[256];
  float m = -1e30f;
  for (int c = threadIdx.x; c < Nc; c += 256) m = fmaxf(m, p[c]);
  red[threadIdx.x] = m;
  __syncthreads();
  for (int s = 128; s; s >>= 1) {
    if (threadIdx.x < s) red[threadIdx.x] = fmaxf(red[threadIdx.x], red[threadIdx.x + s]);
    __syncthreads();
  }
  m = red[0];
  __syncthreads();
  float d = 0.0f;
  for (int c = threadIdx.x; c < Nc; c += 256) d += __expf(p[c] - m);
  red[threadIdx.x] = d;
  __syncthreads();
  for (int s = 128; s; s >>= 1) {
    if (threadIdx.x < s) red[threadIdx.x] += red[threadIdx.x + s];
    __syncthreads();
  }
  float rd = 1.0f / red[0];
  for (int c = threadIdx.x; c < Nc; c += 256)
    out[(size_t)row * Nc + c] = __expf(p[c] - m) * rd;
}

// ------------------------------------ col max of c2c (symmetric -> row max)
__global__ void rowmax_kernel(const float* __restrict__ c2c, float* __restrict__ cm) {
  int wid  = (blockIdx.x * blockDim.x + threadIdx.x) >> 5;
  int lane = threadIdx.x & 31;
  if (wid >= C_NUM) return;
  const float* p = c2c + (size_t)wid * C_NUM;
  float m = -1e30f;
  for (int c = lane; c < C_NUM; c += 32) m = fmaxf(m, p[c]);
  #pragma unroll
  for (int o = 16; o; o >>= 1) m = fmaxf(m, __shfl_down(m, o, 32));
  if (lane == 0) cm[wid] = m;
}

// EexpT[j,i] = exp(c2c[i,j] - colmax[j])  (transposed so A-kernel is X@Y^T)
__global__ void eexpT_kernel(const float* __restrict__ c2c, const float* __restrict__ cm,
                             float* __restrict__ EexpT) {
  int t = blockIdx.x * blockDim.x + threadIdx.x;  // 512*512
  int j = t >> 9, i = t & 511;
  EexpT[t] = __expf(c2c[(size_t)i * C_NUM + j] - cm[j]);
}

// ---------------------------- per-student softmax over answered exercises
// val[n,c] = sum_l Xm[n,l]*softmax_l(S[idx[n,l],c]); active from running max.
// fixed shift exp(w-10) is exact (cancels in num/den); w in [-10,10] or NEG.
__global__ void gather_softmax_kernel(const float* __restrict__ S, int Nc,
                                      const int* __restrict__ eidx,
                                      const unsigned char* __restrict__ mask,
                                      const float* __restrict__ scores,
                                      float* __restrict__ out_val,
                                      float* __restrict__ out_act) {
  int n = blockIdx.x;
  int c = blockIdx.y * blockDim.x + threadIdx.x;
  __shared__ int   sidx[LSEQ];
  __shared__ float sx[LSEQ];
  for (int l = threadIdx.x; l < LSEQ; l += blockDim.x) {
    bool v  = mask[n * LSEQ + l] != 0;
    sidx[l] = v ? eidx[n * LSEQ + l] : -1;
    sx[l]   = v ? scores[n * LSEQ + l] : 0.0f;
  }
  __syncthreads();
  float m = -1e30f, d = 0.0f, num = 0.0f;
  for (int l = 0; l < LSEQ; ++l) {
    int   idx = sidx[l];
    float w   = (idx >= 0) ? S[(size_t)idx * Nc + c] : NEGV;
    m = fmaxf(m, w);
    float e = __expf(w - 10.0f);   // underflows to 0 for NEG entries
    d   += e;
    num += sx[l] * e;
  }
  float val    = (d > 0.0f) ? num / d : 0.0f;
  float active = (m > -1e8f) ? 1.0f : 0.0f;
  if (out_act) {                       // concept branch: store val*active, active
    out_val[(size_t)n * Nc + c] = val * active;
    out_act[(size_t)n * Nc + c] = active;
  } else {                             // potential branch: store B = val
    out_val[(size_t)n * Nc + c] = val;
  }
}

// --------------- A = ((val*active)@Eexp) / (active@Eexp), via EexpT, WMMA
// 16x64 strips: 4 shared B-fragments x {numerator, denominator} = 8 WMMAs
// per K-step on 6 fragment loads
__global__ void A_kernel(const float* __restrict__ valact, const float* __restrict__ act,
                         const float* __restrict__ EexpT, float* __restrict__ Aout) {
  int wid  = (blockIdx.x * blockDim.x + threadIdx.x) >> 5;  // 8 x 8 strips
  int tm = wid >> 3, tn = wid & 7;
  int lane = threadIdx.x & 31;
  int half = lane >> 4, r = lane & 15;
  const float* xv = valact + (size_t)(tm * 16 + r) * C_NUM + 2 * half;
  const float* xa = act    + (size_t)(tm * 16 + r) * C_NUM + 2 * half;
  const float* yb[4];
  #pragma unroll
  for (int j = 0; j < 4; ++j)
    yb[j] = EexpT + (size_t)(tn * 64 + j * 16 + r) * C_NUM + 2 * half;
  v8f aN[4] = {}, aD[4] = {};
  #pragma unroll 4
  for (int k = 0; k < C_NUM; k += 4) {
    v2f a1 = *(const v2f*)(xv + k);
    v2f a2 = *(const v2f*)(xa + k);
    #pragma unroll
    for (int j = 0; j < 4; ++j) {
      v2f b = *(const v2f*)(yb[j] + k);
      aN[j] = wmma4(a1, b, aN[j]);
      aD[j] = wmma4(a2, b, aD[j]);
    }
  }
  int n = lane & 15, mo = half * 8;
  #pragma unroll
  for (int j = 0; j < 4; ++j)
    #pragma unroll
    for (int v = 0; v < 8; ++v)
      Aout[(size_t)(tm * 16 + mo + v) * C_NUM + tn * 64 + j * 16 + n] =
          aN[j][v] / aD[j][v];
}

// ------- Y = mix(A@sm_e2c^T, B@sm_e2p^T) with sigmoid slip/guess, fused
// 16x64 strips, shared A fragment across 4 column tiles
__global__ void Y_kernel(const float* __restrict__ A, const float* __restrict__ Bm,
                         const float* __restrict__ sm_e2c, const float* __restrict__ sm_e2p,
                         const float* __restrict__ lambd, const float* __restrict__ guess,
                         const float* __restrict__ slide, float* __restrict__ Y) {
  int wid  = (blockIdx.x * blockDim.x + threadIdx.x) >> 5;  // 8 x 64 strips
  int tm = wid >> 6, tn = wid & 63;
  int lane = threadIdx.x & 31;
  int half = lane >> 4, r = lane & 15;
  v8f accA[4] = {}, accB[4] = {};
  {
    const float* xa = A + (size_t)(tm * 16 + r) * C_NUM + 2 * half;
    const float* yb[4];
    #pragma unroll
    for (int j = 0; j < 4; ++j)
      yb[j] = sm_e2c + (size_t)(tn * 64 + j * 16 + r) * C_NUM + 2 * half;
    #pragma unroll 4
    for (int k = 0; k < C_NUM; k += 4) {
      v2f a = *(const v2f*)(xa + k);
      #pragma unroll
      for (int j = 0; j < 4; ++j) {
        v2f b = *(const v2f*)(yb[j] + k);
        accA[j] = wmma4(a, b, accA[j]);
      }
    }
  }
  {
    const float* xa = Bm + (size_t)(tm * 16 + r) * P_NUM + 2 * half;
    const float* yb[4];
    #pragma unroll
    for (int j = 0; j < 4; ++j)
      yb[j] = sm_e2p + (size_t)(tn * 64 + j * 16 + r) * P_NUM + 2 * half;
    #pragma unroll 4
    for (int k = 0; k < P_NUM; k += 4) {
      v2f a = *(const v2f*)(xa + k);
      #pragma unroll
      for (int j = 0; j < 4; ++j) {
        v2f b = *(const v2f*)(yb[j] + k);
        accB[j] = wmma4(a, b, accB[j]);
      }
    }
  }
  int n = lane & 15, mo = half * 8;
  #pragma unroll
  for (int j = 0; j < 4; ++j) {
    int col = tn * 64 + j * 16 + n;
    float lam = sigm(lambd[col]);
    float sl  = sigm(slide[col]);
    float gu  = sigm(guess[col]);
    #pragma unroll
    for (int v = 0; v < 8; ++v) {
      float y = (1.0f - lam) * accA[j][v] + lam * accB[j][v];
      y = fminf(fmaxf(y, 1e-8f), 1.0f - 1e-8f);
      y = (1.0f - sl) * y + gu * (1.0f - y);
      Y[(size_t)(tm * 16 + mo + v) * E_NUM + col] = y;
    }
  }
}

// ---------------------------------------------------------------------------
extern "C" void kernel_launch(void* const* d_in, const int* in_sizes, int n_in,
                              void* d_out, int out_size, void* d_ws, size_t ws_size,
                              hipStream_t stream) {
  const int*           exer_idx = (const int*)d_in[0];
  const unsigned char* mask     = (const unsigned char*)d_in[1];
  const float*         scores   = (const float*)d_in[2];
  const float*         exer_m   = (const float*)d_in[3];
  const float*         conc_m   = (const float*)d_in[4];
  const float*         pote_m   = (const float*)d_in[5];
  const float*         lambd    = (const float*)d_in[6];
  const float*         guess    = (const float*)d_in[7];
  const float*         slide    = (const float*)d_in[8];
  const float*         adj      = (const float*)d_in[9];

  float* Aout = (float*)d_out;                 // (128, 512)
  float* Yout = Aout + (size_t)NSTU * C_NUM;   // (128, 4096)

  // workspace carve (~28 MB of f32)
  float* w = (float*)d_ws;
  float* e2c    = w; w += (size_t)E_NUM * C_NUM;
  float* sm_e2c = w; w += (size_t)E_NUM * C_NUM;
  float* e2p    = w; w += (size_t)E_NUM * P_NUM;
  float* sm_e2p = w; w += (size_t)E_NUM * P_NUM;
  float* c2c    = w; w += (size_t)C_NUM * C_NUM;
  float* EexpT  = w; w += (size_t)C_NUM * C_NUM;
  float* cmax   = w; w += C_NUM;
  float* inv_e  = w; w += E_NUM;
  float* inv_c  = w; w += C_NUM;
  float* inv_p  = w; w += P_NUM;
  float* valact = w; w += (size_t)NSTU * C_NUM;
  float* act    = w; w += (size_t)NSTU * C_NUM;
  float* Bmat   = w; w += (size_t)NSTU * P_NUM;

  // 1) inverse row norms
  rnorm_kernel<<<E_NUM / 8, 256, 0, stream>>>(exer_m, inv_e, E_NUM);
  rnorm_kernel<<<C_NUM / 8, 256, 0, stream>>>(conc_m, inv_c, C_NUM);
  rnorm_kernel<<<P_NUM / 8, 256, 0, stream>>>(pote_m, inv_p, P_NUM);

  // 2) cosine similarity matrices (WMMA f32), one wave / 16x64 strip
  sim_kernel<<<(E_NUM / 16) * (C_NUM / 64) / 8, 256, 0, stream>>>(
      exer_m, conc_m, inv_e, inv_c, adj, e2c, E_NUM, C_NUM);
  sim_kernel<<<(C_NUM / 16) * (C_NUM / 64) / 8, 256, 0, stream>>>(
      conc_m, conc_m, inv_c, inv_c, nullptr, c2c, C_NUM, C_NUM);
  sim_kernel<<<(E_NUM / 16) * (P_NUM / 64) / 8, 256, 0, stream>>>(
      exer_m, pote_m, inv_e, inv_p, nullptr, e2p, E_NUM, P_NUM);

  // 3) per-exercise softmax over concepts / potentials
  rowsoftmax_kernel<<<E_NUM, 256, 0, stream>>>(e2c, sm_e2c, C_NUM);
  rowsoftmax_kernel<<<E_NUM, 256, 0, stream>>>(e2p, sm_e2p, P_NUM);

  // 4) Eexp (transposed) with column max (c2c symmetric -> row max)
  rowmax_kernel<<<C_NUM / 8, 256, 0, stream>>>(c2c, cmax);
  eexpT_kernel<<<(C_NUM * C_NUM) / 256, 256, 0, stream>>>(c2c, cmax, EexpT);

  // 5) ragged gather-softmax over answered exercises
  gather_softmax_kernel<<<dim3(NSTU, C_NUM / 256), 256, 0, stream>>>(
      e2c, C_NUM, exer_idx, mask, scores, valact, act);
  gather_softmax_kernel<<<dim3(NSTU, P_NUM / 256), 256, 0, stream>>>(
      e2p, P_NUM, exer_idx, mask, scores, Bmat, nullptr);

  // 6) A = ((val*active)@Eexp)/(active@Eexp)  (WMMA, shared B fragments)
  A_kernel<<<(NSTU / 16) * (C_NUM / 64) / 8, 256, 0, stream>>>(
      valact, act, EexpT, Aout);

  // 7) fused prediction Y (WMMA x2 + sigmoid mixing)
  Y_kernel<<<(NSTU / 16) * (E_NUM / 64) / 8, 256, 0, stream>>>(
      Aout, Bmat, sm_e2c, sm_e2p, lambd, guess, slide, Yout);
}